// ClassicalHopfield_87144886436109
// MI455X (gfx1250) — compile-verified
//
#include <hip/hip_runtime.h>
#include <stdint.h>

#define SEQ 8
#define FDIM 128
#define ROWS_PER_BLOCK 2
#define GUESSES 4
#define BLOCK_THREADS (ROWS_PER_BLOCK * GUESSES * 32)

typedef __attribute__((ext_vector_type(4))) unsigned int v4u;
typedef __attribute__((ext_vector_type(8))) int v8i;
typedef __attribute__((ext_vector_type(4))) int v4i;

// ---------- wave32 butterfly sum (all lanes get the total) ----------
__device__ __forceinline__ float wave_sum(float x) {
  x += __shfl_xor(x, 16, 32);
  x += __shfl_xor(x, 8, 32);
  x += __shfl_xor(x, 4, 32);
  x += __shfl_xor(x, 2, 32);
  x += __shfl_xor(x, 1, 32);
  return x;
}

// ---------- JAX threefry2x32 (exact) ----------
__device__ __forceinline__ uint32_t rotl32(uint32_t x, int r) {
  return (x << r) | (x >> (32 - r));
}

__device__ __forceinline__ void threefry2x32(uint32_t k0, uint32_t k1,
                                             uint32_t& x0, uint32_t& x1) {
  const uint32_t k2 = k0 ^ k1 ^ 0x1BD11BDAu;
  x0 += k0; x1 += k1;
#define TFR(d) { x0 += x1; x1 = rotl32(x1, d); x1 ^= x0; }
  TFR(13) TFR(15) TFR(26) TFR(6)  x0 += k1; x1 += k2 + 1u;
  TFR(17) TFR(29) TFR(16) TFR(24) x0 += k2; x1 += k0 + 2u;
  TFR(13) TFR(15) TFR(26) TFR(6)  x0 += k0; x1 += k1 + 3u;
  TFR(17) TFR(29) TFR(16) TFR(24) x0 += k1; x1 += k2 + 4u;
  TFR(13) TFR(15) TFR(26) TFR(6)  x0 += k2; x1 += k0 + 5u;
#undef TFR
}

// ---------- erfinv, Giles f32 polynomial (matches XLA ErfInv32) ----------
__device__ __forceinline__ float erfinv_f32(float x) {
  float w = -__logf((1.0f - x) * (1.0f + x));
  float p;
  if (w < 5.0f) {
    w -= 2.5f;
    p = 2.81022636e-08f;
    p = fmaf(p, w, 3.43273939e-07f);
    p = fmaf(p, w, -3.5233877e-06f);
    p = fmaf(p, w, -4.39150654e-06f);
    p = fmaf(p, w, 0.00021858087f);
    p = fmaf(p, w, -0.00125372503f);
    p = fmaf(p, w, -0.00417768164f);
    p = fmaf(p, w, 0.246640727f);
    p = fmaf(p, w, 1.50140941f);
  } else {
    w = sqrtf(w) - 3.0f;
    p = -0.000200214257f;
    p = fmaf(p, w, 0.000100950558f);
    p = fmaf(p, w, 0.00134934322f);
    p = fmaf(p, w, -0.00367342844f);
    p = fmaf(p, w, 0.00573950773f);
    p = fmaf(p, w, -0.0076224613f);
    p = fmaf(p, w, 0.00943887047f);
    p = fmaf(p, w, 1.00167406f);
    p = fmaf(p, w, 2.83297682f);
  }
  return p * x;
}

// jax.random.normal(key(1), (3,B,128)) element at flat index `idx`
__device__ __forceinline__ float jax_normal_key1(uint32_t idx, uint32_t half) {
  uint32_t c0, c1;
  const bool lo = idx < half;
  if (lo) { c0 = idx; c1 = idx + half; } else { c0 = idx - half; c1 = idx; }
  threefry2x32(0u, 1u, c0, c1);            // key(1) == {hi=0, lo=1}
  const uint32_t bits = lo ? c0 : c1;
  const float f = __uint_as_float((bits >> 9) | 0x3F800000u) - 1.0f; // [0,1)
  const float u = fmaxf(fmaf(f, 1.99999994f, -0.99999994f), -0.99999994f);
  return 1.41421356237f * erfinv_f32(u);
}

__global__ void __launch_bounds__(BLOCK_THREADS)
hopfield_clean_kernel(const float* __restrict__ embd,
                      const int* __restrict__ n_iter_p,
                      float* __restrict__ out, int B) {
  __shared__ float tile[SEQ * ROWS_PER_BLOCK * FDIM];   // 8 KB

  const int row0 = blockIdx.x * ROWS_PER_BLOCK;
  const int wave = threadIdx.x >> 5;
  const int lane = threadIdx.x & 31;

  // ---------------- TDM async stage: seq_vecs tile -> LDS ----------------
#if __has_builtin(__builtin_amdgcn_tensor_load_to_lds) && __has_builtin(__builtin_amdgcn_s_wait_tensorcnt)
  if (wave == 0) {
    const uint64_t gaddr = (uint64_t)(uintptr_t)(const void*)(embd + (uint64_t)row0 * FDIM);
    const uint32_t lds_addr = (uint32_t)(uintptr_t)(void*)&tile[0]; // low 32b of flat = LDS offset
    const uint64_t s_stride = (uint64_t)B * FDIM;   // z (seq) stride, elements

    v4u g0; v8i g1; v4i g2, g3; v8i g4;
    g0[0] = 1u;                                            // count=1, user descriptor
    g0[1] = lds_addr;                                      // lds_addr
    g0[2] = (uint32_t)(gaddr & 0xFFFFFFFFu);               // global_addr lo
    g0[3] = (uint32_t)((gaddr >> 32) & 0x01FFFFFFu) | (2u << 30); // addr hi | type=2

    g1[0] = (int)(2u << 16);                               // wg_mask=0, data_size=4B
    g1[1] = (int)(((uint32_t)FDIM & 0xFFFFu) << 16);       // tensor_dim0 lo16
    g1[2] = (int)(((uint32_t)B & 0xFFFFu) << 16);          // dim0 hi16=0 | tensor_dim1 lo16
    g1[3] = (int)((((uint32_t)B >> 16) & 0xFFFFu) | ((uint32_t)FDIM << 16)); // dim1 hi | tile_dim0
    g1[4] = (int)(((uint32_t)ROWS_PER_BLOCK) | ((uint32_t)SEQ << 16));       // tile_dim1 | tile_dim2
    g1[5] = (int)FDIM;                                     // tensor_dim0_stride lo32 (row stride)
    g1[6] = (int)(((uint32_t)(s_stride & 0xFFFFu)) << 16); // d0_stride hi16=0 | d1_stride lo16
    g1[7] = (int)((uint32_t)(s_stride >> 16));             // d1_stride bits 47:16

    g2[0] = SEQ;                                           // tensor_dim2 (z extent)
    g2[1] = 0; g2[2] = 0; g2[3] = 0;                       // dim3/dim2_stride/tile_dim3 unused
    g3[0] = 0; g3[1] = 0; g3[2] = 0; g3[3] = 0;
    g4[0] = 0; g4[1] = 0; g4[2] = 0; g4[3] = 0;            // extra group (clang-23 6-arg form)
    g4[4] = 0; g4[5] = 0; g4[6] = 0; g4[7] = 0;

    __builtin_amdgcn_tensor_load_to_lds(g0, g1, g2, g3, g4, 0);
    __builtin_amdgcn_s_wait_tensorcnt(0);
  }
  __syncthreads();
#else
  for (int i = threadIdx.x; i < SEQ * ROWS_PER_BLOCK * FDIM; i += blockDim.x) {
    const int s = i / (ROWS_PER_BLOCK * FDIM);
    const int rem = i % (ROWS_PER_BLOCK * FDIM);
    tile[i] = embd[((uint64_t)s * B + row0 + rem / FDIM) * FDIM + (rem % FDIM)];
  }
  __syncthreads();
#endif

  // ---------------- per-wave task: one (row, guess) ----------------
  const int r = wave >> 2;               // 0..ROWS_PER_BLOCK-1
  const int g = wave & 3;                // 0..3
  const int row = row0 + r;
  const int n_iter = n_iter_p[0];

  // This row's 8 seq vectors: 32 VGPRs, resident across all iterations.
  float vs[SEQ][4];
#pragma unroll
  for (int s = 0; s < SEQ; ++s) {
    const float4 t = *(const float4*)&tile[(s * ROWS_PER_BLOCK + r) * FDIM + lane * 4];
    vs[s][0] = t.x; vs[s][1] = t.y; vs[s][2] = t.z; vs[s][3] = t.w;
  }

  float c[4], m[4], v[4];
  if (g == 0) {
    const float4 t = *(const float4*)&embd[((uint64_t)SEQ * B + row) * FDIM + lane * 4];
    c[0] = t.x; c[1] = t.y; c[2] = t.z; c[3] = t.w;
  } else {
    const uint32_t half = (3u * (uint32_t)B * (uint32_t)FDIM) >> 1;
#pragma unroll
    for (int j = 0; j < 4; ++j) {
      const uint32_t idx = ((uint32_t)(g - 1) * (uint32_t)B + (uint32_t)row) * FDIM
                         + (uint32_t)(lane * 4 + j);
      c[j] = jax_normal_key1(idx, half);
    }
  }
#pragma unroll
  for (int j = 0; j < 4; ++j) { m[j] = 0.0f; v[j] = 0.0f; }

  float cn[4] = {0.0f, 0.0f, 0.0f, 0.0f};
  float b1p = 1.0f, b2p = 1.0f;

  for (int t = 1; t <= n_iter; ++t) {
    // cn = c / max(||c||, 1e-12)
    float ss = c[0]*c[0] + c[1]*c[1] + c[2]*c[2] + c[3]*c[3];
    ss = wave_sum(ss);
    const float nrm = fmaxf(sqrtf(ss), 1e-12f);
    const float inv = 1.0f / nrm;
#pragma unroll
    for (int j = 0; j < 4; ++j) cn[j] = c[j] * inv;
    if (t == n_iter) break;   // scan returns cn at start of last iter

    // d_s = cn . v_s  (8 simultaneous wave reductions)
    float d[SEQ];
#pragma unroll
    for (int s = 0; s < SEQ; ++s) {
      float p = cn[0]*vs[s][0] + cn[1]*vs[s][1] + cn[2]*vs[s][2] + cn[3]*vs[s][3];
      d[s] = wave_sum(p);
    }

    // u = -2 * sum_s d_s * v_s ;  grad = (u - (cn.u) cn) / nrm
    float u[4] = {0.0f, 0.0f, 0.0f, 0.0f};
#pragma unroll
    for (int s = 0; s < SEQ; ++s)
#pragma unroll
      for (int j = 0; j < 4; ++j) u[j] = fmaf(d[s], vs[s][j], u[j]);
#pragma unroll
    for (int j = 0; j < 4; ++j) u[j] *= -2.0f;

    float cu = cn[0]*u[0] + cn[1]*u[1] + cn[2]*u[2] + cn[3]*u[3];
    cu = wave_sum(cu);

    // Adam step (B1=0.9, B2=0.999, LR=0.01, EPS=1e-8), bias-corrected
    b1p *= 0.9f; b2p *= 0.999f;
    const float mc = 1.0f / (1.0f - b1p);
    const float vc = 1.0f / (1.0f - b2p);
#pragma unroll
    for (int j = 0; j < 4; ++j) {
      const float gr = (u[j] - cu * cn[j]) * inv;
      m[j] = fmaf(0.9f, m[j], 0.1f * gr);
      v[j] = fmaf(0.999f, v[j], 0.001f * gr * gr);
      const float mh = m[j] * mc;
      const float vh = v[j] * vc;
      c[j] -= 0.01f * mh / (sqrtf(vh) + 1e-8f);
    }
  }

  // out[row, g*128 + f]
  float4 o; o.x = cn[0]; o.y = cn[1]; o.z = cn[2]; o.w = cn[3];
  *(float4*)&out[(uint64_t)row * (GUESSES * FDIM) + g * FDIM + lane * 4] = o;
}

extern "C" void kernel_launch(void* const* d_in, const int* in_sizes, int n_in,
                              void* d_out, int out_size, void* d_ws, size_t ws_size,
                              hipStream_t stream) {
  const float* embd = (const float*)d_in[0];
  const int* n_iter = (const int*)d_in[1];
  float* out = (float*)d_out;
  const int B = in_sizes[0] / ((SEQ + 1) * FDIM);       // 16384
  const int blocks = (B + ROWS_PER_BLOCK - 1) / ROWS_PER_BLOCK;
  hipLaunchKernelGGL(hopfield_clean_kernel, dim3(blocks), dim3(BLOCK_THREADS), 0, stream,
                     embd, n_iter, out, B);
}